// classifier_selective_ts_90099823936041
// MI455X (gfx1250) — compile-verified
//
#include <hip/hip_runtime.h>
#include <hip/hip_bf16.h>

typedef __attribute__((ext_vector_type(2))) float v2f;
typedef __attribute__((ext_vector_type(8))) float v8f;

#define BS   8192
#define MBR  19
#define DIM  768
#define DHALF 384
#define XCOLS (MBR * DIM)   // 14592

// ---------------------------------------------------------------------------
// fea_u[b][d] = sum_m x[b][m*768+d]   (float4 vectorized, pure HBM stream)
// ---------------------------------------------------------------------------
__global__ void sum_branches_kernel(const float* __restrict__ x,
                                    float* __restrict__ fea_u) {
    int idx = blockIdx.x * 256 + threadIdx.x;        // over BS*192 float4s
    int b  = idx / 192;
    int d4 = idx % 192;
    const float4* p = (const float4*)x + (size_t)b * (XCOLS / 4) + d4;
    float4 s = make_float4(0.f, 0.f, 0.f, 0.f);
#pragma unroll
    for (int m = 0; m < MBR; ++m) {
        float4 v = p[m * (DIM / 4)];
        s.x += v.x; s.y += v.y; s.z += v.z; s.w += v.w;
    }
    ((float4*)fea_u)[idx] = s;
}

// ---------------------------------------------------------------------------
// Generic fp32 WMMA "NT" GEMM: C[M,N] = A[M,K] * B[N,K]^T + bias, optional ReLU
// One wave computes one 16x16 tile via V_WMMA_F32_16X16X4_F32.
// A-frag: lane L: row=L%16, holds K = k0 + (L/16)*2 .. +1  (2 VGPRs)
// B-frag: lane L: col=L%16, holds K = k0 + (L/16)*2 .. +1
// C: VGPR e -> row = e + (L>=16 ? 8 : 0), col = L%16
// ---------------------------------------------------------------------------
template <int K, int LDA, int LDB, int NTILES, int NCOLS, int LDC, bool RELU>
__global__ void gemm_nt_kernel(const float* __restrict__ A,
                               const float* __restrict__ B,
                               const float* __restrict__ bias,
                               float* __restrict__ C) {
    int wave = blockIdx.x * (blockDim.x >> 5) + (threadIdx.x >> 5);
    int mt = wave / NTILES;
    int nt = wave % NTILES;
    int lane = threadIdx.x & 31;
    int r  = lane & 15;
    int kb = (lane >> 4) << 1;     // 0 or 2

    const float* Ap = A + (size_t)(mt * 16 + r) * LDA + kb;
    const float* Bp = B + (size_t)(nt * 16 + r) * LDB + kb;

    v8f acc = {};
#pragma unroll 4
    for (int k = 0; k < K; k += 4) {
        v2f a = *(const v2f*)(Ap + k);
        v2f b = *(const v2f*)(Bp + k);
        acc = __builtin_amdgcn_wmma_f32_16x16x4_f32(false, a, false, b,
                                                    (short)0, acc, false, false);
    }

    int col = nt * 16 + r;
    float bv = bias[col];                       // padded when NCOLS%16 != 0
    int rowbase = mt * 16 + ((lane >> 4) << 3);
#pragma unroll
    for (int e = 0; e < 8; ++e) {
        float v = acc[e] + bv;
        if (RELU) v = fmaxf(v, 0.f);
        if (NCOLS % 16 != 0) {
            if (col < NCOLS) C[(size_t)(rowbase + e) * LDC + col] = v;
        } else {
            C[(size_t)(rowbase + e) * LDC + col] = v;
        }
    }
}

// ---------------------------------------------------------------------------
// Fused: att logits (WMMA) -> online softmax over 19 branches -> fea_v.
// Block = 8 waves: 2 batch-subtiles x 4 out-subtiles => 32x64 tile.
// fea_z tile (32x384 = 48KB) staged in LDS once, reused for all 19 branches:
// hot loop is 1 ds_load_b64 (A) + 1 global_load_b64 (W_br) per WMMA.
// att is never materialized; x is read exactly once here (online softmax).
// ---------------------------------------------------------------------------
__global__ void att_softmax_feav_kernel(const float* __restrict__ x,
                                        const float* __restrict__ fea_z,
                                        const float* __restrict__ W_br,
                                        const float* __restrict__ b_br,
                                        float* __restrict__ fea_v) {
    __shared__ float As[32 * DHALF];            // 48 KB
    int tid = threadIdx.x;
    int bt = blockIdx.x / 12;                   // 256 batch tiles of 32
    int ot = blockIdx.x % 12;                   // 12 out tiles of 64
    int b0blk = bt * 32;

    // cooperative stage of fea_z[b0blk .. b0blk+32) -> LDS (float4)
    {
        const float4* src = (const float4*)(fea_z + (size_t)b0blk * DHALF);
        float4* dst = (float4*)As;
#pragma unroll
        for (int i = 0; i < (32 * DHALF / 4) / 256; ++i)   // 12 iters
            dst[tid + i * 256] = src[tid + i * 256];
    }
    __syncthreads();

    int wid  = tid >> 5;
    int lane = tid & 31;
    int wb = wid >> 2, wo = wid & 3;
    int b0 = b0blk + wb * 16;
    int o0 = ot * 64 + wo * 16;
    int r  = lane & 15;
    int kb = (lane >> 4) << 1;
    int hi = lane >> 4;

    const float* Ap = As + (wb * 16 + r) * DHALF + kb;     // LDS
    int col = o0 + r;

    const float* xp[8];
#pragma unroll
    for (int e = 0; e < 8; ++e)
        xp[e] = x + (size_t)(b0 + hi * 8 + e) * XCOLS + col;

    float rm[8], rs[8], rn[8];
#pragma unroll
    for (int e = 0; e < 8; ++e) { rm[e] = -3.4e38f; rs[e] = 0.f; rn[e] = 0.f; }

#pragma unroll 1
    for (int m = 0; m < MBR; ++m) {
        const float* Bp = W_br + ((size_t)m * DIM + o0 + r) * DHALF + kb;
        v8f acc = {};
#pragma unroll 4
        for (int k = 0; k < DHALF; k += 4) {
            v2f a = *(const v2f*)(Ap + k);
            v2f b = *(const v2f*)(Bp + k);
            acc = __builtin_amdgcn_wmma_f32_16x16x4_f32(false, a, false, b,
                                                        (short)0, acc, false, false);
        }
        float bb = b_br[m * DIM + col];
#pragma unroll
        for (int e = 0; e < 8; ++e) {
            float lg = acc[e] + bb;
            float xv = xp[e][m * DIM];
            float nm = fmaxf(rm[e], lg);
            float c  = __expf(rm[e] - nm);
            float p  = __expf(lg - nm);
            rs[e] = rs[e] * c + p;
            rn[e] = rn[e] * c + p * xv;
            rm[e] = nm;
        }
    }
#pragma unroll
    for (int e = 0; e < 8; ++e)
        fea_v[(size_t)(b0 + hi * 8 + e) * DIM + col] = rn[e] / rs[e];
}

// ---------------------------------------------------------------------------
// Column partial sums / sum-of-squares (deterministic: no atomics).
// 64 blocks x 128 rows. Writes partials[part][NF].
// ---------------------------------------------------------------------------
template <int NF>
__global__ void colred_part_kernel(const float* __restrict__ h,
                                   float* __restrict__ ps,
                                   float* __restrict__ pss) {
    constexpr int R = 256 / NF;        // rowgroups per block
    constexpr int ROWS = BS / 64;      // 128 rows per block
    int col = threadIdx.x % NF;
    int rg  = threadIdx.x / NF;
    int r0  = blockIdx.x * ROWS + rg * (ROWS / R);
    float s = 0.f, ss = 0.f;
    for (int rr = r0; rr < r0 + ROWS / R; ++rr) {
        float v = h[(size_t)rr * NF + col];
        s += v; ss += v * v;
    }
    int part = blockIdx.x * R + rg;
    ps[part * NF + col]  = s;
    pss[part * NF + col] = ss;
}

// ---------------------------------------------------------------------------
// Finalize BN stats and fold the affine BN into the next layer's weights:
//   W'[o][i] = W[o][i] * gamma[i]*rsqrt(var+eps)
//   b'[o]    = b[o] + sum_i W[o][i] * (beta[i] - mean[i]*gamma[i]*rsqrt(var+eps))
// Rows [NOUT, NOUT_PAD) are zero-padded so downstream WMMA tiles are unguarded.
// ---------------------------------------------------------------------------
template <int NIN, int PARTS, int NOUT, int NOUT_PAD>
__global__ void fold_bn_kernel(const float* __restrict__ ps,
                               const float* __restrict__ pss,
                               const float* __restrict__ g,
                               const float* __restrict__ be,
                               const float* __restrict__ W,
                               const float* __restrict__ bvec,
                               float* __restrict__ Wp,
                               float* __restrict__ bp) {
    __shared__ float scale[NIN];
    __shared__ float shift[NIN];
    int t = threadIdx.x;
    if (t < NIN) {
        float s = 0.f, ss = 0.f;
        for (int k = 0; k < PARTS; ++k) {
            s  += ps[k * NIN + t];
            ss += pss[k * NIN + t];
        }
        float mu  = s * (1.f / BS);
        float var = ss * (1.f / BS) - mu * mu;     // biased variance (jnp.var)
        float rsq = rsqrtf(var + 1e-5f);
        float sc  = g[t] * rsq;
        scale[t] = sc;
        shift[t] = be[t] - mu * sc;
    }
    __syncthreads();
    if (t < NOUT_PAD) {
        if (t < NOUT) {
            float acc = bvec[t];
            for (int i = 0; i < NIN; ++i) {
                float w = W[(size_t)t * NIN + i];
                Wp[(size_t)t * NIN + i] = w * scale[i];
                acc += w * shift[i];
            }
            bp[t] = acc;
        } else {
            for (int i = 0; i < NIN; ++i) Wp[(size_t)t * NIN + i] = 0.f;
            bp[t] = 0.f;
        }
    }
}

// ---------------------------------------------------------------------------
extern "C" void kernel_launch(void* const* d_in, const int* in_sizes, int n_in,
                              void* d_out, int out_size, void* d_ws, size_t ws_size,
                              hipStream_t stream) {
    const float* x    = (const float*)d_in[0];
    const float* W_fc = (const float*)d_in[1];
    const float* b_fc = (const float*)d_in[2];
    const float* W_br = (const float*)d_in[3];
    const float* b_br = (const float*)d_in[4];
    const float* W1   = (const float*)d_in[5];
    const float* b1   = (const float*)d_in[6];
    const float* g1   = (const float*)d_in[7];
    const float* be1  = (const float*)d_in[8];
    const float* W2   = (const float*)d_in[9];
    const float* b2   = (const float*)d_in[10];
    const float* g2   = (const float*)d_in[11];
    const float* be2  = (const float*)d_in[12];
    const float* W3   = (const float*)d_in[13];
    const float* b3   = (const float*)d_in[14];
    float* out = (float*)d_out;

    // workspace layout (floats), with lifetime-based aliasing (~42 MB total):
    //  region A: fea_u  (dead after fz-GEMM)  -> reused as fea_v
    //  region B: fea_z  (dead after att)      -> reused as h1
    float* w       = (float*)d_ws;
    float* regionA = w;                        // 8192*768
    float* regionB = regionA + BS * DIM;       // 8192*384
    float* h2      = regionB + BS * DHALF;     // 8192*128
    float* ps1     = h2 + BS * 128;            // 64*256
    float* pq1     = ps1 + 64 * 256;           // 64*256
    float* ps2     = pq1 + 64 * 256;           // 128*128
    float* pq2     = ps2 + 128 * 128;          // 128*128
    float* W2p     = pq2 + 128 * 128;          // 128*256
    float* b2p     = W2p + 128 * 256;          // 128
    float* W3p     = b2p + 128;                // 160*128 (zero-padded rows)
    float* b3p     = W3p + 160 * 128;          // 160

    float* fea_u = regionA;
    float* fea_v = regionA;                    // alias (fea_u dead)
    float* fea_z = regionB;
    float* h1    = regionB;                    // alias (fea_z dead)

    // 1) fea_u = sum over 19 branches          (HBM-bound pass 1 over x)
    sum_branches_kernel<<<BS * (DIM / 4) / 256, 256, 0, stream>>>(x, fea_u);

    // 2) fea_z = fea_u @ W_fc^T + b_fc         [8192,768]x[384,768]^T
    gemm_nt_kernel<768, 768, 768, 24, 384, 384, false>
        <<<(BS / 16) * 24 / 8, 256, 0, stream>>>(fea_u, W_fc, b_fc, fea_z);

    // 3) fused att-GEMM + online softmax + weighted sum (pass 2 over x)
    att_softmax_feav_kernel<<<(BS / 32) * (DIM / 64), 256, 0, stream>>>(
        x, fea_z, W_br, b_br, fea_v);

    // 4) h1 = relu(fea_v @ W1^T + b1)
    gemm_nt_kernel<768, 768, 768, 16, 256, 256, true>
        <<<(BS / 16) * 16 / 8, 256, 0, stream>>>(fea_v, W1, b1, h1);

    // 5) BN1 stats (deterministic partials) + fold into (W2,b2)
    colred_part_kernel<256><<<64, 256, 0, stream>>>(h1, ps1, pq1);
    fold_bn_kernel<256, 64, 128, 128><<<1, 256, 0, stream>>>(
        ps1, pq1, g1, be1, W2, b2, W2p, b2p);

    // 6) h2 = relu(h1 @ W2'^T + b2')
    gemm_nt_kernel<256, 256, 256, 8, 128, 128, true>
        <<<(BS / 16) * 8 / 8, 256, 0, stream>>>(h1, W2p, b2p, h2);

    // 7) BN2 stats + fold into (W3,b3), zero-padded to 160 rows
    colred_part_kernel<128><<<64, 256, 0, stream>>>(h2, ps2, pq2);
    fold_bn_kernel<128, 128, 150, 160><<<1, 256, 0, stream>>>(
        ps2, pq2, g2, be2, W3, b3, W3p, b3p);

    // 8) out = h2 @ W3'^T + b3'   (N=150, store-guarded)
    gemm_nt_kernel<128, 128, 128, 10, 150, 150, false>
        <<<(BS / 16) * 10 / 8, 256, 0, stream>>>(h2, W3p, b3p, out);
}